// HGT_27590869910181
// MI455X (gfx1250) — compile-verified
//
#include <hip/hip_runtime.h>
#include <math.h>

// ---------------------------------------------------------------- constants
#define NB_N 100000
#define NG_N 50000
#define E_BB 800000
#define E_GB 200000
#define E_BG 200000
#define INV_SQRT_D 0.17677669529663687f   // 1/sqrt(32)

typedef float v2f __attribute__((ext_vector_type(2)));
typedef float v8f __attribute__((ext_vector_type(8)));

// ---------------------------------------------------------------- helpers
__device__ __forceinline__ float gelu_tanh(float x) {
  float x3 = x * x * x;
  return 0.5f * x * (1.0f + tanhf(0.7978845608028654f * (x + 0.044715f * x3)));
}

__device__ __forceinline__ void atomicMaxF(float* addr, float val) {
  unsigned int* ai = (unsigned int*)addr;
  unsigned int old = *ai;
  while (true) {
    float f = __uint_as_float(old);
    if (f >= val) break;
    unsigned int assumed = old;
    old = atomicCAS(ai, assumed, __float_as_uint(val));
    if (old == assumed) break;
  }
}

// ---------------------------------------------------------------- GEMM (WMMA f32 16x16x4)
// Y[n,64] = act(X[n,64] @ W[64,64] + b)   ; optional gated skip:
//   a = sigmoid(*skipp); Y = a*Y + (1-a)*resid
// block = 128 threads (4 waves); each wave: 16 rows x 64 cols (64 WMMAs)
__global__ __launch_bounds__(128)
void gemm64_kernel(const float* __restrict__ X, const float* __restrict__ Wg,
                   const float* __restrict__ Bg, float* __restrict__ Y,
                   int nrows, int act, const float* __restrict__ skipp,
                   const float* __restrict__ resid) {
  __shared__ float Wl[64 * 64];
  __shared__ float Bl[64];
  int t = threadIdx.x;
  for (int i = t; i < 4096; i += 128) Wl[i] = Wg[i];
  if (t < 64) Bl[t] = Bg[t];
  __syncthreads();

  int wave = t >> 5, lane = t & 31;
  int g = lane >> 4, ln = lane & 15;
  int r0 = blockIdx.x * 64 + wave * 16;

  // A fragments: 16x4 f32 tile layout -> lane holds (row=ln, K = k0+2g, k0+2g+1)
  int ar = r0 + ln;
  if (ar >= nrows) ar = nrows - 1;            // clamp (EXEC must stay full for WMMA)
  const float* xr = X + (size_t)ar * 64 + 2 * g;
  v2f a[16];
#pragma unroll
  for (int kt = 0; kt < 16; ++kt) {
    a[kt].x = xr[kt * 4 + 0];
    a[kt].y = xr[kt * 4 + 1];
  }

  float gate = 0.0f;
  if (skipp) gate = 1.0f / (1.0f + expf(-skipp[0]));

  for (int nt = 0; nt < 4; ++nt) {
    int n0 = nt * 16;
    v8f acc = {0.f, 0.f, 0.f, 0.f, 0.f, 0.f, 0.f, 0.f};
#pragma unroll
    for (int kt = 0; kt < 16; ++kt) {
      int kb = kt * 4 + 2 * g;                 // B 4x16 tile: lane holds (K=kb,kb+1; col=ln)
      v2f b;
      b.x = Wl[(kb + 0) * 64 + n0 + ln];
      b.y = Wl[(kb + 1) * 64 + n0 + ln];
      acc = __builtin_amdgcn_wmma_f32_16x16x4_f32(false, a[kt], false, b,
                                                  (short)0, acc, false, false);
    }
    int col = n0 + ln;
#pragma unroll
    for (int r = 0; r < 8; ++r) {
      int row = r0 + r + 8 * g;                // C layout: VGPR r -> M=r (g=0), M=r+8 (g=1)
      if (row < nrows) {
        float v = acc[r] + Bl[col];
        if (act == 1) v = fmaxf(v, 0.0f);
        if (skipp) {
          float res = resid[(size_t)row * 64 + col];
          v = gate * v + (1.0f - gate) * res;
        }
        Y[(size_t)row * 64 + col] = v;
      }
    }
  }
}

// ---------------------------------------------------------------- fold a_rel/m_rel into K/V weights
// Wc[i, h*32+e] = sum_d W[i, h*32+d] * rel[h,d,e] ; bc likewise from b.
__global__ void relcomb_kernel(const float* __restrict__ W, const float* __restrict__ b,
                               const float* __restrict__ rel, float* __restrict__ Wc,
                               float* __restrict__ bc) {
  int tid = blockIdx.x * blockDim.x + threadIdx.x;
  if (tid < 4096) {
    int i = tid >> 6, he = tid & 63;
    int h = he >> 5, e = he & 31;
    float acc = 0.f;
#pragma unroll
    for (int d = 0; d < 32; ++d) acc += W[i * 64 + h * 32 + d] * rel[h * 1024 + d * 32 + e];
    Wc[i * 64 + he] = acc;
  } else if (tid < 4160) {
    int he = tid - 4096;
    int h = he >> 5, e = he & 31;
    float acc = 0.f;
#pragma unroll
    for (int d = 0; d < 32; ++d) acc += b[h * 32 + d] * rel[h * 1024 + d * 32 + e];
    bc[he] = acc;
  }
}

// ---------------------------------------------------------------- edge kernels
// one wave per (edge, head): coalesced 32-lane loads, shfl reduction
__global__ void edge_logits_kernel(const float* __restrict__ q, const float* __restrict__ krel,
                                   const int* __restrict__ si, const int* __restrict__ di,
                                   const float* __restrict__ p_rel, int E,
                                   float* __restrict__ logit, float* __restrict__ m) {
  long long tid = (long long)blockIdx.x * blockDim.x + threadIdx.x;
  long long w = tid >> 5;
  int lane = (int)(tid & 31);
  if (w >= (long long)E * 2) return;           // wave-uniform exit
  int e = (int)(w >> 1), h = (int)(w & 1);
  int s = si[e], d = di[e];
  float prod = q[(size_t)d * 64 + h * 32 + lane] * krel[(size_t)s * 64 + h * 32 + lane];
#pragma unroll
  for (int off = 16; off > 0; off >>= 1) prod += __shfl_xor(prod, off, 32);
  if (lane == 0) {
    float lg = prod * p_rel[h] * INV_SQRT_D;
    logit[(size_t)e * 2 + h] = lg;
    atomicMaxF(&m[(size_t)d * 2 + h], lg);
  }
}

__global__ void edge_exp_kernel(float* __restrict__ eb, const float* __restrict__ m,
                                const int* __restrict__ di, float* __restrict__ ssum, int E) {
  long long tid = (long long)blockIdx.x * blockDim.x + threadIdx.x;
  if (tid >= (long long)E * 2) return;
  int e = (int)(tid >> 1), h = (int)(tid & 1);
  int d = di[e];
  float ex = expf(eb[(size_t)e * 2 + h] - m[(size_t)d * 2 + h]);
  eb[(size_t)e * 2 + h] = ex;
  atomicAdd(&ssum[(size_t)d * 2 + h], ex);
}

// 64 lanes per edge (coalesced vrel read + coalesced atomicAdd on agg)
__global__ void edge_scatter_kernel(const float* __restrict__ eb, const float* __restrict__ ssum,
                                    const float* __restrict__ vrel, const int* __restrict__ si,
                                    const int* __restrict__ di, float* __restrict__ agg, int E) {
  long long tid = (long long)blockIdx.x * blockDim.x + threadIdx.x;
  if (tid >= (long long)E * 64) return;
  int e = (int)(tid >> 6), c = (int)(tid & 63), h = c >> 5;
  int s = si[e], d = di[e];
  float al = eb[(size_t)e * 2 + h] / (ssum[(size_t)d * 2 + h] + 1e-16f);
  atomicAdd(&agg[(size_t)d * 64 + c], al * vrel[(size_t)s * 64 + c]);
}

// ---------------------------------------------------------------- misc elementwise
__global__ void fill_kernel(float* __restrict__ p, float v, long long n) {
  long long i = (long long)blockIdx.x * blockDim.x + threadIdx.x;
  if (i < n) p[i] = v;
}

__global__ void min_gelu2_kernel(const float* __restrict__ a, const float* __restrict__ b,
                                 float* __restrict__ o, long long n) {
  long long i = (long long)blockIdx.x * blockDim.x + threadIdx.x;
  if (i < n) o[i] = gelu_tanh(fminf(a[i], b[i]));
}

__global__ void gelu1_kernel(const float* __restrict__ a, float* __restrict__ o, long long n) {
  long long i = (long long)blockIdx.x * blockDim.x + threadIdx.x;
  if (i < n) o[i] = gelu_tanh(a[i]);
}

__global__ void matvec_kernel(const float* __restrict__ X, const float* __restrict__ w,
                              const float* __restrict__ b, float* __restrict__ y, int n) {
  int i = blockIdx.x * blockDim.x + threadIdx.x;
  if (i >= n) return;
  const float4* xr = (const float4*)(X + (size_t)i * 64);
  const float4* wr = (const float4*)w;
  float s = b[0];
#pragma unroll
  for (int c = 0; c < 16; ++c) {
    float4 xv = xr[c], wv = wr[c];
    s += xv.x * wv.x + xv.y * wv.y + xv.z * wv.z + xv.w * wv.w;
  }
  y[i] = s;
}

// ---------------------------------------------------------------- launch
extern "C" void kernel_launch(void* const* d_in, const int* in_sizes, int n_in,
                              void* d_out, int out_size, void* d_ws, size_t ws_size,
                              hipStream_t stream) {
  (void)in_sizes; (void)n_in; (void)out_size; (void)ws_size;
  // ---- input mapping (params flattened in pytree order: dict keys sorted) ----
  // 0:x_bus 1:x_gen ; 2..55 convs[0],convs[1] (27 each):
  //  +0 a.bus.b +1 a.bus.w +2 a.gen.b +3 a.gen.w
  //  +4 k.bus.b +5 k.bus.w +6 k.gen.b +7 k.gen.w
  //  +8 q.bus.b +9 q.bus.w +10 q.gen.b +11 q.gen.w
  //  +12 rel.bb.a_rel +13 .m_rel +14 .p_rel ; +15..17 rel.bg ; +18..20 rel.gb
  //  +21 skip.bus +22 skip.gen ; +23 v.bus.b +24 v.bus.w +25 v.gen.b +26 v.gen.w
  // 56..59 lin.bus.b/w lin.gen.b/w ; 60..65 mlp0.b/w mlp1.b/w mlp2.b/w
  // 66 ei_bb 67 ei_gb 68 ei_bg
  const float* x_bus = (const float*)d_in[0];
  const float* x_gen = (const float*)d_in[1];
  auto P = [&](int i) { return (const float*)d_in[i]; };
  const int* ei_bb = (const int*)d_in[66];
  const int* ei_gb = (const int*)d_in[67];
  const int* ei_bg = (const int*)d_in[68];

  // ---- workspace layout ----
  float* ws = (float*)d_ws;
  size_t off = 0;
  auto A = [&](size_t n) { float* p = ws + off; off += n; return p; };
  const size_t NB64 = (size_t)NB_N * 64, NG64 = (size_t)NG_N * 64;
  float* buf_bus0 = A(NB64);  float* buf_gen0 = A(NG64);
  float* buf_bus1 = A(NB64);  float* buf_gen1 = A(NG64);
  float* q_bus = A(NB64);     float* q_gen = A(NG64);
  float* krel = A(NB64);      float* vrel = A(NB64);   // reused as gelu bufs post-edge
  float* agg_bb = A(NB64);    float* agg_gb = A(NB64); float* agg_bg = A(NG64);
  float* eb = A((size_t)E_BB * 2);
  float* mbuf = A((size_t)NB_N * 2);
  float* sbuf = A((size_t)NB_N * 2);
  float* Wck = A(64 * 64);    float* bck = A(64);
  float* Wcv = A(64 * 64);    float* bcv = A(64);

  auto blks = [](long long n, int b) { return (unsigned)((n + b - 1) / b); };
  auto gemm = [&](const float* X, const float* W, const float* B, float* Y, int n,
                  int act, const float* sk, const float* rs) {
    gemm64_kernel<<<dim3(blks(n, 64)), 128, 0, stream>>>(X, W, B, Y, n, act, sk, rs);
  };
  auto fill = [&](float* p, float v, long long n) {
    fill_kernel<<<dim3(blks(n, 256)), 256, 0, stream>>>(p, v, n);
  };

  // ---- input projections + relu ----
  gemm(x_bus, P(57), P(56), buf_bus0, NB_N, 1, nullptr, nullptr);
  gemm(x_gen, P(59), P(58), buf_gen0, NG_N, 1, nullptr, nullptr);

  float* cb = buf_bus0; float* cg = buf_gen0;
  float* nb = buf_bus1; float* ng = buf_gen1;

  for (int L = 0; L < 2; ++L) {
    int base = 2 + L * 27;
    // Q projections
    gemm(cb, P(base + 9), P(base + 8), q_bus, NB_N, 0, nullptr, nullptr);
    gemm(cg, P(base + 11), P(base + 10), q_gen, NG_N, 0, nullptr, nullptr);
    // zero aggregates
    fill(agg_bb, 0.f, (long long)NB64);
    fill(agg_gb, 0.f, (long long)NB64);
    fill(agg_bg, 0.f, (long long)NG64);

    struct Et {
      const int *si, *di; int E; const float* src; int nsrc, ndst;
      int kb, kw, vb, vw, rel; float* agg; const float* qd;
    };
    Et ets[3] = {
      { ei_bb, ei_bb + E_BB, E_BB, cb, NB_N, NB_N, base + 4, base + 5, base + 23, base + 24, base + 12, agg_bb, q_bus },
      { ei_gb, ei_gb + E_GB, E_GB, cg, NG_N, NB_N, base + 6, base + 7, base + 25, base + 26, base + 18, agg_gb, q_bus },
      { ei_bg, ei_bg + E_BG, E_BG, cb, NB_N, NG_N, base + 4, base + 5, base + 23, base + 24, base + 15, agg_bg, q_gen },
    };

    for (int i = 0; i < 3; ++i) {
      Et& t = ets[i];
      relcomb_kernel<<<dim3(17), 256, 0, stream>>>(P(t.kw), P(t.kb), P(t.rel + 0), Wck, bck);
      relcomb_kernel<<<dim3(17), 256, 0, stream>>>(P(t.vw), P(t.vb), P(t.rel + 1), Wcv, bcv);
      gemm(t.src, Wck, bck, krel, t.nsrc, 0, nullptr, nullptr);
      gemm(t.src, Wcv, bcv, vrel, t.nsrc, 0, nullptr, nullptr);
      fill(mbuf, -3.402823466e38f, (long long)t.ndst * 2);
      fill(sbuf, 0.f, (long long)t.ndst * 2);
      edge_logits_kernel<<<dim3(blks((long long)t.E * 64, 256)), 256, 0, stream>>>(
          t.qd, krel, t.si, t.di, P(t.rel + 2), t.E, eb, mbuf);
      edge_exp_kernel<<<dim3(blks((long long)t.E * 2, 256)), 256, 0, stream>>>(
          eb, mbuf, t.di, sbuf, t.E);
      edge_scatter_kernel<<<dim3(blks((long long)t.E * 64, 256)), 256, 0, stream>>>(
          eb, sbuf, vrel, t.si, t.di, t.agg, t.E);
    }

    // min across edge types -> gelu, then a_lin with fused sigmoid-gated skip
    min_gelu2_kernel<<<dim3(blks((long long)NB64, 256)), 256, 0, stream>>>(agg_bb, agg_gb, krel, (long long)NB64);
    gelu1_kernel<<<dim3(blks((long long)NG64, 256)), 256, 0, stream>>>(agg_bg, vrel, (long long)NG64);
    gemm(krel, P(base + 1), P(base + 0), nb, NB_N, 0, P(base + 21), cb);
    gemm(vrel, P(base + 3), P(base + 2), ng, NG_N, 0, P(base + 22), cg);

    float* tmp;
    tmp = cb; cb = nb; nb = tmp;
    tmp = cg; cg = ng; ng = tmp;
  }

  // ---- MLP head on bus ----
  gemm(cb, P(61), P(60), q_bus, NB_N, 1, nullptr, nullptr);
  gemm(q_bus, P(63), P(62), krel, NB_N, 1, nullptr, nullptr);
  matvec_kernel<<<dim3(blks(NB_N, 256)), 256, 0, stream>>>(krel, P(65), P(64), (float*)d_out, NB_N);
}